// HeteroGraphormerLayerComplete_16045997817898
// MI455X (gfx1250) — compile-verified
//
#include <hip/hip_runtime.h>
#include <hip/hip_bf16.h>
#include <math.h>

typedef __attribute__((ext_vector_type(2))) float v2f;
typedef __attribute__((ext_vector_type(8))) float v8f;

#define CCH 128   // channels
#define NH  4     // heads
#define HD  32    // head dim

// ---- order-preserving float <-> uint encoding for atomicMax-based segment max ----
__device__ __forceinline__ unsigned enc_f32(float f) {
    unsigned u = __float_as_uint(f);
    return (u & 0x80000000u) ? ~u : (u | 0x80000000u);
}
__device__ __forceinline__ float dec_f32(unsigned e) {
    return (e & 0x80000000u) ? __uint_as_float(e & 0x7FFFFFFFu) : __uint_as_float(~e);
}
#define ENC_NEG_INF 0x007FFFFFu   // enc_f32(-inf)

// ------------------------------- fills -------------------------------
__global__ void fill_f32_kernel(float* __restrict__ p, float v, int n) {
    int i = blockIdx.x * blockDim.x + threadIdx.x;
    if (i < n) p[i] = v;
}
__global__ void fill_u32_kernel(unsigned* __restrict__ p, unsigned v, int n) {
    int i = blockIdx.x * blockDim.x + threadIdx.x;
    if (i < n) p[i] = v;
}

// --------------------- fused Q/K/V projection (fp32 WMMA) ---------------------
// out{Q,K,V}[n, j] = sum_k x[n,k] * W{q,k,v}[j,k] + b{q,k,v}[j]   (x @ W.T + b)
// Block: 256 threads = 8 waves. Block owns 16 node rows (staged in LDS);
// wave w owns output cols [16w, 16w+16) and computes Q, K, V tiles sharing
// the same A fragment. K-loop: 32 steps of v_wmma_f32_16x16x4_f32 x3.
__global__ __launch_bounds__(256) void qkv_wmma_kernel(
    const float* __restrict__ x,
    const float* __restrict__ Wq, const float* __restrict__ bq,
    const float* __restrict__ Wk, const float* __restrict__ bk,
    const float* __restrict__ Wv, const float* __restrict__ bv,
    float* __restrict__ Q, float* __restrict__ K, float* __restrict__ V)
{
    __shared__ float xt[16 * CCH];
    const int rowBase = blockIdx.x * 16;

    // cooperative 16x128 fp32 tile load (8 KB), float4 per thread x2
    for (int i = threadIdx.x; i < 16 * CCH / 4; i += 256) {
        ((float4*)xt)[i] = ((const float4*)(x + (size_t)rowBase * CCH))[i];
    }
    __syncthreads();

    const int wave = threadIdx.x >> 5;
    const int lane = threadIdx.x & 31;
    const int colBase = wave * 16;
    const int col  = colBase + (lane & 15);
    const int half = lane >> 4;          // A layout: lanes 0-15 -> K0/K1, 16-31 -> K2/K3
    const int rowA = lane & 15;

    v8f cq = {}; v8f ck = {}; v8f cv = {};
    for (int k = 0; k < CCH; k += 4) {
        const int kk = k + half * 2;
        v2f a  = *(const v2f*)&xt[rowA * CCH + kk];
        v2f bq2 = *(const v2f*)&Wq[(size_t)col * CCH + kk];  // B[k][j] = W[j][k]
        v2f bk2 = *(const v2f*)&Wk[(size_t)col * CCH + kk];
        v2f bv2 = *(const v2f*)&Wv[(size_t)col * CCH + kk];
        cq = __builtin_amdgcn_wmma_f32_16x16x4_f32(false, a, false, bq2, (short)0, cq, false, false);
        ck = __builtin_amdgcn_wmma_f32_16x16x4_f32(false, a, false, bk2, (short)0, ck, false, false);
        cv = __builtin_amdgcn_wmma_f32_16x16x4_f32(false, a, false, bv2, (short)0, cv, false, false);
    }

    const float biasq = bq[col], biask = bk[col], biasv = bv[col];
    const int mOff = (lane < 16) ? 0 : 8;   // C/D layout: VGPR r -> M = r (+8 for hi lanes)
#pragma unroll
    for (int r = 0; r < 8; ++r) {
        const size_t o = (size_t)(rowBase + mOff + r) * CCH + col;
        Q[o] = cq[r] + biasq;
        K[o] = ck[r] + biask;
        V[o] = cv[r] + biasv;
    }
}

// --------------------- per-edge attention scores + segment max + degrees ---------------------
__global__ __launch_bounds__(256) void edge_score_kernel(
    const float* __restrict__ Qd, const float* __restrict__ Ks,
    const int* __restrict__ src, const int* __restrict__ dst,
    const float* __restrict__ sb, const float* __restrict__ etype_bias,
    float* __restrict__ scores, unsigned* __restrict__ menc,
    float* __restrict__ degS, float* __restrict__ degD, int nE)
{
    const int e    = blockIdx.x * 8 + (threadIdx.x >> 5);
    const int lane = threadIdx.x & 31;
    if (e >= nE) return;
    const int s = src[e], d = dst[e];

    const float4 q = ((const float4*)(Qd + (size_t)d * CCH))[lane];
    const float4 k = ((const float4*)(Ks + (size_t)s * CCH))[lane];
    float p = q.x * k.x + q.y * k.y + q.z * k.z + q.w * k.w;
    // reduce within the 8 lanes of each head (lane covers channels [4*lane,4*lane+4))
    p += __shfl_xor(p, 1);
    p += __shfl_xor(p, 2);
    p += __shfl_xor(p, 4);

    if ((lane & 7) == 0) {
        const int h = lane >> 3;
        const float score = p * 0.17677669529663687f /* 1/sqrt(32) */ + sb[e] + etype_bias[0];
        scores[(size_t)e * NH + h] = score;
        atomicMax(&menc[d * NH + h], enc_f32(score));
    }
    if (lane == 0) {   // total degree: out-deg of src + in-deg of dst
        atomicAdd(&degS[s], 1.0f);
        atomicAdd(&degD[d], 1.0f);
    }
}

// --------------------- exp(score - m[dst]) + segment sum ---------------------
__global__ void edge_exp_kernel(const int* __restrict__ dst,
                                const unsigned* __restrict__ menc,
                                float* __restrict__ scores,
                                float* __restrict__ denom, int nE)
{
    const int idx = blockIdx.x * blockDim.x + threadIdx.x;
    if (idx >= nE * NH) return;
    const int e = idx >> 2, h = idx & 3;
    const int d = dst[e];
    const float m  = dec_f32(menc[d * NH + h]);
    const float ex = __expf(scores[idx] - m);
    scores[idx] = ex;
    atomicAdd(&denom[d * NH + h], ex);
}

// --------------------- weighted scatter-add of V[src] into out[dst] ---------------------
__global__ __launch_bounds__(256) void edge_scatter_kernel(
    const float* __restrict__ Vs, const int* __restrict__ src, const int* __restrict__ dst,
    const float* __restrict__ ex, const float* __restrict__ denom,
    float* __restrict__ out, int nE)
{
    const int e    = blockIdx.x * 8 + (threadIdx.x >> 5);
    const int lane = threadIdx.x & 31;
    if (e >= nE) return;
    const int s = src[e], d = dst[e];
    const int h = lane >> 3;
    const float w = ex[(size_t)e * NH + h] / denom[d * NH + h];
    const float4 v = ((const float4*)(Vs + (size_t)s * CCH))[lane];
    float* o = out + (size_t)d * CCH + lane * 4;
    atomicAdd(o + 0, v.x * w);
    atomicAdd(o + 1, v.y * w);
    atomicAdd(o + 2, v.z * w);
    atomicAdd(o + 3, v.w * w);
}

// --------------------- deg + residual + LayerNorm (in-place on d_out) ---------------------
__global__ __launch_bounds__(256) void finalize_ln_kernel(
    float* __restrict__ out, const float* __restrict__ x,
    const float* __restrict__ deg, const float* __restrict__ gamma,
    const float* __restrict__ beta, int nN)
{
    const int n    = blockIdx.x * 8 + (threadIdx.x >> 5);
    const int lane = threadIdx.x & 31;
    if (n >= nN) return;
    const float dg = deg[n];
    const float4 a  = ((const float4*)(out + (size_t)n * CCH))[lane];
    const float4 xv = ((const float4*)(x   + (size_t)n * CCH))[lane];
    float4 t = make_float4(a.x + dg + xv.x, a.y + dg + xv.y,
                           a.z + dg + xv.z, a.w + dg + xv.w);

    float sum = t.x + t.y + t.z + t.w;
#pragma unroll
    for (int off = 16; off >= 1; off >>= 1) sum += __shfl_xor(sum, off);
    const float mu = sum * (1.0f / CCH);

    const float4 dd = make_float4(t.x - mu, t.y - mu, t.z - mu, t.w - mu);
    float vs = dd.x * dd.x + dd.y * dd.y + dd.z * dd.z + dd.w * dd.w;
#pragma unroll
    for (int off = 16; off >= 1; off >>= 1) vs += __shfl_xor(vs, off);
    const float inv = rsqrtf(vs * (1.0f / CCH) + 1e-5f);

    const float4 g = ((const float4*)gamma)[lane];
    const float4 b = ((const float4*)beta)[lane];
    float4 o;
    o.x = dd.x * inv * g.x + b.x;
    o.y = dd.y * inv * g.y + b.y;
    o.z = dd.z * inv * g.z + b.z;
    o.w = dd.w * inv * g.w + b.w;
    ((float4*)(out + (size_t)n * CCH))[lane] = o;
}

// ------------------------------- launcher -------------------------------
extern "C" void kernel_launch(void* const* d_in, const int* in_sizes, int n_in,
                              void* d_out, int out_size, void* d_ws, size_t ws_size,
                              hipStream_t stream) {
    const float* x_A   = (const float*)d_in[0];
    const float* x_B   = (const float*)d_in[1];
    const float* Wq    = (const float*)d_in[2];
    const float* bq    = (const float*)d_in[3];
    const float* Wk    = (const float*)d_in[4];
    const float* bk    = (const float*)d_in[5];
    const float* Wv    = (const float*)d_in[6];
    const float* bv    = (const float*)d_in[7];
    const float* gamma = (const float*)d_in[8];
    const float* beta  = (const float*)d_in[9];
    const float* bias_AB = (const float*)d_in[10];
    const float* bias_BA = (const float*)d_in[11];
    const float* sb_AB   = (const float*)d_in[12];
    const float* sb_BA   = (const float*)d_in[13];
    const int* ei_AB = (const int*)d_in[14];
    const int* ei_BA = (const int*)d_in[15];

    const int N = in_sizes[0] / CCH;   // 100000 (divisible by 16)
    const int E = in_sizes[12];        // 1000000
    const int* src_ab = ei_AB;       const int* dst_ab = ei_AB + E;   // A -> B
    const int* src_ba = ei_BA;       const int* dst_ba = ei_BA + E;   // B -> A

    float* outA = (float*)d_out;
    float* outB = outA + (size_t)N * CCH;

    // ---- workspace carve-up (fp32 elements) ----
    float* ws = (float*)d_ws;
    size_t off = 0;
    float* QA = ws + off; off += (size_t)N * CCH;
    float* KA = ws + off; off += (size_t)N * CCH;
    float* VA = ws + off; off += (size_t)N * CCH;
    float* QB = ws + off; off += (size_t)N * CCH;
    float* KB = ws + off; off += (size_t)N * CCH;
    float* VB = ws + off; off += (size_t)N * CCH;
    float* scAB = ws + off; off += (size_t)E * NH;
    float* scBA = ws + off; off += (size_t)E * NH;
    unsigned* mB   = (unsigned*)(ws + off); off += (size_t)N * NH;
    unsigned* mA   = (unsigned*)(ws + off); off += (size_t)N * NH;
    float*    denB = ws + off; off += (size_t)N * NH;
    float*    denA = ws + off; off += (size_t)N * NH;
    float*    degA = ws + off; off += (size_t)N;
    float*    degB = ws + off; off += (size_t)N;

    const int T = 256;
    // ---- init (must run every call: harness does not re-zero between replays) ----
    {
        const int nOut = 2 * N * CCH;
        fill_f32_kernel<<<(nOut + T - 1) / T, T, 0, stream>>>((float*)d_out, 0.0f, nOut);
        const int nM = N * NH;
        fill_u32_kernel<<<(nM + T - 1) / T, T, 0, stream>>>(mA, ENC_NEG_INF, nM);
        fill_u32_kernel<<<(nM + T - 1) / T, T, 0, stream>>>(mB, ENC_NEG_INF, nM);
        fill_f32_kernel<<<(nM + T - 1) / T, T, 0, stream>>>(denA, 0.0f, nM);
        fill_f32_kernel<<<(nM + T - 1) / T, T, 0, stream>>>(denB, 0.0f, nM);
        fill_f32_kernel<<<(N + T - 1) / T, T, 0, stream>>>(degA, 0.0f, N);
        fill_f32_kernel<<<(N + T - 1) / T, T, 0, stream>>>(degB, 0.0f, N);
    }

    // ---- fused Q/K/V projections (fp32 WMMA), one launch per node type ----
    qkv_wmma_kernel<<<N / 16, T, 0, stream>>>(x_A, Wq, bq, Wk, bk, Wv, bv, QA, KA, VA);
    qkv_wmma_kernel<<<N / 16, T, 0, stream>>>(x_B, Wq, bq, Wk, bk, Wv, bv, QB, KB, VB);

    // ---- edge attention, A->B (Q from B, K/V from A) and B->A ----
    const int eBlocks = (E + 7) / 8;
    edge_score_kernel<<<eBlocks, T, 0, stream>>>(QB, KA, src_ab, dst_ab, sb_AB, bias_AB,
                                                 scAB, mB, degA, degB, E);
    edge_score_kernel<<<eBlocks, T, 0, stream>>>(QA, KB, src_ba, dst_ba, sb_BA, bias_BA,
                                                 scBA, mA, degB, degA, E);

    const int nEH = E * NH;
    edge_exp_kernel<<<(nEH + T - 1) / T, T, 0, stream>>>(dst_ab, mB, scAB, denB, E);
    edge_exp_kernel<<<(nEH + T - 1) / T, T, 0, stream>>>(dst_ba, mA, scBA, denA, E);

    edge_scatter_kernel<<<eBlocks, T, 0, stream>>>(VA, src_ab, dst_ab, scAB, denB, outB, E);
    edge_scatter_kernel<<<eBlocks, T, 0, stream>>>(VB, src_ba, dst_ba, scBA, denA, outA, E);

    // ---- deg + residual + LayerNorm ----
    const int nBlocks = (N + 7) / 8;
    finalize_ln_kernel<<<nBlocks, T, 0, stream>>>(outA, x_A, degA, gamma, beta, N);
    finalize_ln_kernel<<<nBlocks, T, 0, stream>>>(outB, x_B, degB, gamma, beta, N);
}